// GAT_3143916061300
// MI455X (gfx1250) — compile-verified
//
#include <hip/hip_runtime.h>
#include <hip/hip_bf16.h>
#include <math.h>

// ---------------------------------------------------------------------------
// GATv2 (2 layers) for MI455X / gfx1250.
// GEMMs: V_WMMA_F32_16X16X4_F32 (fp32-precise; flops are trivial here).
// Edge phase: wave-per-edge gather/scatter, L2-resident (192MB L2 >> 154MB WS).
// ---------------------------------------------------------------------------

typedef __attribute__((ext_vector_type(2))) float v2f;
typedef __attribute__((ext_vector_type(8))) float v8f;

#define DIN 128
#define DH  32
#define NH  8
#define HD  256   // NH * DH

__device__ __forceinline__ void atomicMaxF(float* addr, float v) {
  // IEEE-754 ordered compare via integer atomics:
  //  - non-negative floats: signed-int max
  //  - negative floats:     unsigned-int min (bit order reversed)
  if (v >= 0.0f)
    atomicMax(reinterpret_cast<int*>(addr), __float_as_int(v));
  else
    atomicMin(reinterpret_cast<unsigned int*>(addr),
              static_cast<unsigned int>(__float_as_int(v)));
}

__device__ __forceinline__ float waveReduceAdd(float v) {
#pragma unroll
  for (int off = 16; off; off >>= 1) v += __shfl_xor(v, off, 32);
  return v;
}

// ------------------------------ utility kernels ----------------------------

__global__ void fill_kernel(float* __restrict__ p, float v, long n) {
  long i = (long)blockIdx.x * blockDim.x + threadIdx.x;
  long stride = (long)gridDim.x * blockDim.x;
  for (; i < n; i += stride) p[i] = v;
}

__global__ void mean_ew_kernel(const float* __restrict__ ew, float* __restrict__ out,
                               int E) {
  int lane = threadIdx.x & 31;
  long i = (long)blockIdx.x * blockDim.x + threadIdx.x;
  long stride = (long)gridDim.x * blockDim.x;
  float s = 0.f;
  for (; i < E; i += stride) s += ew[i];
  s = waveReduceAdd(s);
  if (lane == 0) atomicAdd(out, s / (float)E);
}

// ------------------------------ WMMA GEMM ----------------------------------
// C[N x M] = A[N x K] @ W[K x M], row-major fp32.
// One wave computes TWO adjacent 16x16 tiles (32 output cols) per tn row-tile,
// reusing the A fragment across both B fragments.
// fp32 WMMA fragment layout (ISA 7.12.2, 16x4 A):
//   lane l<16 : row M=l, k in {k0, k0+1};  lane l>=16 : row M=l-16, k in {k0+2, k0+3}

__global__ void wmma_gemm_f32(const float* __restrict__ A, const float* __restrict__ W,
                              float* __restrict__ C, int N, int K, int M) {
  int wave = (int)((blockIdx.x * blockDim.x + threadIdx.x) >> 5);
  int lane = threadIdx.x & 31;
  int pairsM = M >> 5;                        // column-tile pairs
  int tilesN = (N + 15) >> 4;
  if (wave >= tilesN * pairsM) return;        // wave-uniform exit
  int tp = wave % pairsM;
  int tn = wave / pairsM;

  int arow = tn * 16 + (lane & 15);
  if (arow >= N) arow = N - 1;                // clamp loads, keep EXEC full
  int khalf = (lane >> 4) << 1;               // 0 or 2
  int bcol0 = tp * 32 + (lane & 15);
  int bcol1 = bcol0 + 16;

  const float* Arow = A + (size_t)arow * K;
  v8f acc0 = {};
  v8f acc1 = {};
  for (int k = 0; k < K; k += 4) {
    v2f a, b0, b1;
    a.x = Arow[k + khalf];
    a.y = Arow[k + khalf + 1];
    const float* Wk = W + (size_t)(k + khalf) * M;
    b0.x = Wk[bcol0];
    b0.y = Wk[M + bcol0];
    b1.x = Wk[bcol1];
    b1.y = Wk[M + bcol1];
    acc0 = __builtin_amdgcn_wmma_f32_16x16x4_f32(false, a, false, b0, (short)0, acc0,
                                                 false, false);
    acc1 = __builtin_amdgcn_wmma_f32_16x16x4_f32(false, a, false, b1, (short)0, acc1,
                                                 false, false);
  }
  // C/D layout: VGPR r -> row r (lanes 0-15) / row r+8 (lanes 16-31), col = lane&15
  int srow = tn * 16 + ((lane >> 4) << 3);
  int scol0 = tp * 32 + (lane & 15);
#pragma unroll
  for (int r = 0; r < 8; ++r) {
    int row = srow + r;
    if (row < N) {
      C[(size_t)row * M + scol0] = acc0[r];
      C[(size_t)row * M + scol0 + 16] = acc1[r];
    }
  }
}

// ------------------------------ layer 1 edge phase -------------------------
// s[e,h] = sum_c lrelu(xl[src,h,c] + xr[dst,h,c] + w*We[h,c]) * att[h,c]
// one wave per edge; lane = c (0..31); loop over 8 heads.

__global__ void edge_score1(const float* __restrict__ xl, const float* __restrict__ xr,
                            const int* __restrict__ srcA, const int* __restrict__ dstA,
                            const float* __restrict__ ew, const float* __restrict__ meanEw,
                            const float* __restrict__ We, const float* __restrict__ att,
                            float* __restrict__ sbuf, float* __restrict__ mbuf,
                            int E, int N) {
  int e = blockIdx.x * (blockDim.x >> 5) + (threadIdx.x >> 5);
  int lane = threadIdx.x & 31;
  int Eall = E + N;
  if (e >= Eall) return;
  int src, dst; float w;
  if (e < E) { src = srcA[e]; dst = dstA[e]; w = ew[e]; }
  else       { src = e - E;  dst = src;      w = *meanEw; }
  const float* xls = xl + (size_t)src * HD;
  const float* xrd = xr + (size_t)dst * HD;
#pragma unroll
  for (int h = 0; h < NH; ++h) {
    int idx = h * 32 + lane;
    float z = xls[idx] + xrd[idx] + w * We[idx];
    z = (z > 0.f) ? z : 0.2f * z;
    float p = waveReduceAdd(z * att[idx]);
    if (lane == 0) {
      sbuf[(size_t)e * NH + h] = p;
      atomicMaxF(&mbuf[(size_t)dst * NH + h], p);
    }
  }
}

// out1[dst,:] += exp(s - m[dst]) * xl[src,:] ; den[dst,h] += exp(...)
__global__ void edge_agg1(const float* __restrict__ xl,
                          const int* __restrict__ srcA, const int* __restrict__ dstA,
                          const float* __restrict__ sbuf, const float* __restrict__ mbuf,
                          float* __restrict__ den, float* __restrict__ out1,
                          int E, int N) {
  int e = blockIdx.x * (blockDim.x >> 5) + (threadIdx.x >> 5);
  int lane = threadIdx.x & 31;
  int Eall = E + N;
  if (e >= Eall) return;
  int src, dst;
  if (e < E) { src = srcA[e]; dst = dstA[e]; }
  else       { src = e - E;  dst = src; }
  const float* xls = xl + (size_t)src * HD;
  float* od = out1 + (size_t)dst * HD;
#pragma unroll
  for (int h = 0; h < NH; ++h) {
    float ex = expf(sbuf[(size_t)e * NH + h] - mbuf[(size_t)dst * NH + h]);
    if (lane == 0) atomicAdd(&den[(size_t)dst * NH + h], ex);
    atomicAdd(&od[h * 32 + lane], ex * xls[h * 32 + lane]);
  }
}

// h1 = relu(out1/den + b1), in-place on out1
__global__ void finalize1(float* __restrict__ h1, const float* __restrict__ den,
                          const float* __restrict__ b1, int N) {
  long i = (long)blockIdx.x * blockDim.x + threadIdx.x;
  if (i >= (long)N * HD) return;
  int n = (int)(i >> 8);
  int hc = (int)(i & 255);
  float v = h1[i] / (den[(size_t)n * NH + (hc >> 5)] + 1e-16f) + b1[hc];
  h1[i] = v > 0.f ? v : 0.f;
}

// ------------------------------ layer 2 edge phase (H=1, C=32) -------------

__global__ void edge_score2(const float* __restrict__ xl, const float* __restrict__ xr,
                            const int* __restrict__ srcA, const int* __restrict__ dstA,
                            const float* __restrict__ ew, const float* __restrict__ meanEw,
                            const float* __restrict__ We, const float* __restrict__ att,
                            float* __restrict__ sbuf, float* __restrict__ mbuf,
                            int E, int N) {
  int e = blockIdx.x * (blockDim.x >> 5) + (threadIdx.x >> 5);
  int lane = threadIdx.x & 31;
  int Eall = E + N;
  if (e >= Eall) return;
  int src, dst; float w;
  if (e < E) { src = srcA[e]; dst = dstA[e]; w = ew[e]; }
  else       { src = e - E;  dst = src;      w = *meanEw; }
  float z = xl[(size_t)src * DH + lane] + xr[(size_t)dst * DH + lane] + w * We[lane];
  z = (z > 0.f) ? z : 0.2f * z;
  float p = waveReduceAdd(z * att[lane]);
  if (lane == 0) {
    sbuf[e] = p;
    atomicMaxF(&mbuf[dst], p);
  }
}

__global__ void edge_agg2(const float* __restrict__ xl,
                          const int* __restrict__ srcA, const int* __restrict__ dstA,
                          const float* __restrict__ sbuf, const float* __restrict__ mbuf,
                          float* __restrict__ den, float* __restrict__ out2,
                          int E, int N) {
  int e = blockIdx.x * (blockDim.x >> 5) + (threadIdx.x >> 5);
  int lane = threadIdx.x & 31;
  int Eall = E + N;
  if (e >= Eall) return;
  int src, dst;
  if (e < E) { src = srcA[e]; dst = dstA[e]; }
  else       { src = e - E;  dst = src; }
  float ex = expf(sbuf[e] - mbuf[dst]);
  if (lane == 0) atomicAdd(&den[dst], ex);
  atomicAdd(&out2[(size_t)dst * DH + lane], ex * xl[(size_t)src * DH + lane]);
}

// feat = relu(out2/den + b2); pooled per-graph sums + counts
__global__ void finalize2_pool(const float* __restrict__ out2, const float* __restrict__ den2,
                               const float* __restrict__ b2, const int* __restrict__ batch,
                               float* __restrict__ pool, float* __restrict__ cnt, int N) {
  long i = (long)blockIdx.x * blockDim.x + threadIdx.x;
  if (i >= (long)N * DH) return;
  int n = (int)(i >> 5);
  int c = (int)(i & 31);
  float v = out2[i] / (den2[n] + 1e-16f) + b2[c];
  v = v > 0.f ? v : 0.f;
  int g = batch[n];
  atomicAdd(&pool[(size_t)g * DH + c], v);
  if (c == 0) atomicAdd(&cnt[g], 1.0f);
}

// out[b] = sigmoid(mean) @ Wfc + bfc ; one wave per graph
__global__ void final_out_kernel(const float* __restrict__ pool, const float* __restrict__ cnt,
                                 const float* __restrict__ Wfc, const float* __restrict__ bfc,
                                 float* __restrict__ out, int B) {
  int b = blockIdx.x * (blockDim.x >> 5) + (threadIdx.x >> 5);
  int lane = threadIdx.x & 31;
  if (b >= B) return;
  float mean = pool[(size_t)b * DH + lane] / fmaxf(cnt[b], 1.0f);
  float p = 1.0f / (1.0f + expf(-mean));
  float t = waveReduceAdd(p * Wfc[lane]);
  if (lane == 0) out[b] = t + bfc[0];
}

// ------------------------------ launcher -----------------------------------

extern "C" void kernel_launch(void* const* d_in, const int* in_sizes, int n_in,
                              void* d_out, int out_size, void* d_ws, size_t ws_size,
                              hipStream_t stream) {
  (void)n_in; (void)ws_size;
  const float* x    = (const float*)d_in[0];
  const int*   eidx = (const int*)d_in[1];
  const int*   batch= (const int*)d_in[2];
  const float* ew   = (const float*)d_in[3];
  const float* Wl1  = (const float*)d_in[4];
  const float* Wr1  = (const float*)d_in[5];
  const float* We1  = (const float*)d_in[6];
  const float* att1 = (const float*)d_in[7];
  const float* b1   = (const float*)d_in[8];
  const float* Wl2  = (const float*)d_in[9];
  const float* Wr2  = (const float*)d_in[10];
  const float* We2  = (const float*)d_in[11];
  const float* att2 = (const float*)d_in[12];
  const float* b2   = (const float*)d_in[13];
  const float* Wfc  = (const float*)d_in[14];
  const float* bfc  = (const float*)d_in[15];

  const int N = in_sizes[0] / DIN;
  const int E = in_sizes[3];
  const int B = out_size;
  const int Eall = E + N;
  const int* srcA = eidx;       // edge_index[0]
  const int* dstA = eidx + E;   // edge_index[1]

  float* ws = (float*)d_ws;
  size_t off = 0;
  float* xl1  = ws + off; off += (size_t)N * HD;
  float* xr1  = ws + off; off += (size_t)N * HD;
  float* out1 = ws + off; off += (size_t)N * HD;    // becomes h1 in-place
  float* s1   = ws + off; off += (size_t)Eall * NH;
  float* m1   = ws + off; off += (size_t)N * NH;
  float* den1 = ws + off; off += (size_t)N * NH;
  float* xl2  = ws + off; off += (size_t)N * DH;
  float* xr2  = ws + off; off += (size_t)N * DH;
  float* out2 = ws + off; off += (size_t)N * DH;
  float* s2   = ws + off; off += (size_t)Eall;
  float* m2   = ws + off; off += (size_t)N;
  float* den2 = ws + off; off += (size_t)N;
  float* pool = ws + off; off += (size_t)B * DH;
  float* cnt  = ws + off; off += (size_t)B;
  float* meanEw = ws + off; off += 1;

  const int TB = 256;                 // 8 waves per block
  const float NEG_INF = -__builtin_inff();
  auto cdiv = [](long a, long b) { return (int)((a + b - 1) / b); };

  // init
  fill_kernel<<<1024, TB, 0, stream>>>(out1, 0.f, (long)N * HD);
  fill_kernel<<<256, TB, 0, stream>>>(m1, NEG_INF, (long)N * NH);
  fill_kernel<<<256, TB, 0, stream>>>(den1, 0.f, (long)N * NH);
  fill_kernel<<<256, TB, 0, stream>>>(out2, 0.f, (long)N * DH);
  fill_kernel<<<64, TB, 0, stream>>>(m2, NEG_INF, (long)N);
  fill_kernel<<<64, TB, 0, stream>>>(den2, 0.f, (long)N);
  fill_kernel<<<8, TB, 0, stream>>>(pool, 0.f, (long)B * DH);
  fill_kernel<<<1, TB, 0, stream>>>(cnt, 0.f, (long)B);
  fill_kernel<<<1, 32, 0, stream>>>(meanEw, 0.f, 1);
  mean_ew_kernel<<<512, TB, 0, stream>>>(ew, meanEw, E);

  // layer 1 GEMMs (WMMA fp32): [N x 128] @ [128 x 256]
  {
    int waves = ((N + 15) >> 4) * (HD >> 5);
    int blocks = cdiv(waves, TB / 32);
    wmma_gemm_f32<<<blocks, TB, 0, stream>>>(x, Wl1, xl1, N, DIN, HD);
    wmma_gemm_f32<<<blocks, TB, 0, stream>>>(x, Wr1, xr1, N, DIN, HD);
  }

  int eblocks = cdiv(Eall, TB / 32);
  edge_score1<<<eblocks, TB, 0, stream>>>(xl1, xr1, srcA, dstA, ew, meanEw,
                                          We1, att1, s1, m1, E, N);
  edge_agg1<<<eblocks, TB, 0, stream>>>(xl1, srcA, dstA, s1, m1, den1, out1, E, N);
  finalize1<<<cdiv((long)N * HD, TB), TB, 0, stream>>>(out1, den1, b1, N);

  // layer 2 GEMMs: [N x 256] @ [256 x 32]
  {
    int waves = ((N + 15) >> 4) * (DH >> 5);
    int blocks = cdiv(waves, TB / 32);
    wmma_gemm_f32<<<blocks, TB, 0, stream>>>(out1, Wl2, xl2, N, HD, DH);
    wmma_gemm_f32<<<blocks, TB, 0, stream>>>(out1, Wr2, xr2, N, HD, DH);
  }

  edge_score2<<<eblocks, TB, 0, stream>>>(xl2, xr2, srcA, dstA, ew, meanEw,
                                          We2, att2, s2, m2, E, N);
  edge_agg2<<<eblocks, TB, 0, stream>>>(xl2, srcA, dstA, s2, m2, den2, out2, E, N);
  finalize2_pool<<<cdiv((long)N * DH, TB), TB, 0, stream>>>(out2, den2, b2, batch,
                                                            pool, cnt, N);
  final_out_kernel<<<cdiv(B, TB / 32), TB, 0, stream>>>(pool, cnt, Wfc, bfc,
                                                        (float*)d_out, B);
}